// FocusOther_50474455662621
// MI455X (gfx1250) — compile-verified
//
#include <hip/hip_runtime.h>

// FocusOther (space-to-depth, stride 2): x (32, 3, 1024, 1024) fp32
//   -> out (32, 12, 512, 512), channel groups in order:
//   g0 = x[...,0::2,0::2], g1 = x[...,1::2,0::2],
//   g2 = x[...,0::2,1::2], g3 = x[...,1::2,1::2]
//
// Bandwidth-bound: 768 MiB total -> ~34.6 us floor at 23.3 TB/s.
// Each thread: 8 consecutive input floats (2x global_load_b128 NT),
// deinterleave into 4 even-col + 4 odd-col, 2x global_store_b128 NT.
// blockIdx.y carries (b*C + c) so the div/mod by 3 is uniform SALU work;
// per-thread VALU is just a few shifts/ands + address MADs.

typedef float v4f __attribute__((ext_vector_type(4)));

#define FB   32
#define FC   3
#define FH   1024
#define FW   1024
#define FH2  (FH / 2)
#define FW2  (FW / 2)
#define PLANE      (FH2 * FW2)           // 262144 floats per output plane
#define CHUNKS_PER_ROW (FW / 8)          // 128 chunks (threads) per input row
#define THREADS_PER_IMG (FH * CHUNKS_PER_ROW)  // 131072 threads per (b,c) image

__global__ __launch_bounds__(256) void
focus_s2d_kernel(const float* __restrict__ in, float* __restrict__ out)
{
    // Uniform (SALU) part: which (b, c) image this block works on.
    const int bc = blockIdx.y;           // b*FC + c, 0..95
    const int c  = bc % FC;              // uniform -> scalar division
    const int b  = bc / FC;

    // Per-thread part: position within the 1024x1024 image.
    const int t     = blockIdx.x * 256 + threadIdx.x;   // 0..131071 exactly
    const int chunk = t & (CHUNKS_PER_ROW - 1);         // 8-float chunk in row
    const int h     = t >> 7;                           // input row 0..1023
    const int i     = h >> 1;                           // output row
    const int r     = h & 1;                            // row parity

    // ---- 2x b128 NT loads: 8 consecutive floats of row h ----
    const v4f* ip = reinterpret_cast<const v4f*>(
        in + ((size_t)bc * FH + h) * FW + (size_t)chunk * 8);
    v4f a = __builtin_nontemporal_load(ip);
    v4f d = __builtin_nontemporal_load(ip + 1);

    // Deinterleave: even columns / odd columns.
    v4f even = (v4f){a.x, a.z, d.x, d.z};
    v4f odd  = (v4f){a.y, a.w, d.y, d.w};

    // Output plane: (b*12 + r*3 + c) for even cols, +6 planes for odd cols.
    //   scalar base = (b*12 + c) * PLANE ; per-thread adds r*3*PLANE + i*W2 + chunk*4
    const size_t base = (size_t)(b * 12 + c) * PLANE;
    const size_t off  = base + (size_t)r * (3 * PLANE)
                             + (size_t)i * FW2 + (size_t)chunk * 4;

    // ---- 2x b128 NT stores, contiguous across lanes ----
    __builtin_nontemporal_store(even, reinterpret_cast<v4f*>(out + off));
    __builtin_nontemporal_store(odd,  reinterpret_cast<v4f*>(out + off + 6 * PLANE));
}

extern "C" void kernel_launch(void* const* d_in, const int* in_sizes, int n_in,
                              void* d_out, int out_size, void* d_ws, size_t ws_size,
                              hipStream_t stream)
{
    (void)in_sizes; (void)n_in; (void)out_size; (void)d_ws; (void)ws_size;
    const float* x = reinterpret_cast<const float*>(d_in[0]);
    float* y       = reinterpret_cast<float*>(d_out);

    dim3 block(256);                              // 8 wave32s
    dim3 grid(THREADS_PER_IMG / 256, FB * FC);    // (512, 96) — exact cover, no tail
    focus_s2d_kernel<<<grid, block, 0, stream>>>(x, y);
}